// PointNetFeaturePropagation_39195871543463
// MI455X (gfx1250) — compile-verified
//
#include <hip/hip_runtime.h>
#include <hip/hip_bf16.h>

typedef __attribute__((ext_vector_type(2))) float v2f;
typedef __attribute__((ext_vector_type(8))) float v8f;

#define NB    8
#define NPT   8192
#define SFEW  2048
#define DFEW  256
#define DLRG  128
#define CIN   384
#define C1    256
#define C2    128
#define NT    65536           // NB*NPT total columns
#define EPS_BN 1e-5f
#define EPS_W  1e-8f

#define TS1 (CIN * 16 + 16)   // padded sub-tile stride (dwords), GEMM1
#define TS2 (C1  * 16 + 16)   // padded sub-tile stride (dwords), GEMM2

// ---------------- zero stats ----------------
__global__ void k_zero(float* __restrict__ p, int n) {
  int i = blockIdx.x * 256 + threadIdx.x;
  if (i < n) p[i] = 0.0f;
}

// ---------------- 3-NN interpolation -> X rows [128..383] ----------------
__global__ __launch_bounds__(256) void k_interp(
    const float* __restrict__ xyz_few, const float* __restrict__ feat_few,
    const float* __restrict__ xyz_large, float* __restrict__ X) {
  __shared__ float fx[SFEW], fy[SFEW], fz[SFEW];
  int b = blockIdx.y;
  int n = blockIdx.x * 256 + threadIdx.x;
  const float* xf = xyz_few + (size_t)b * SFEW * 3;
  for (int i = threadIdx.x; i < SFEW; i += 256) {
    fx[i] = xf[i * 3 + 0];
    fy[i] = xf[i * 3 + 1];
    fz[i] = xf[i * 3 + 2];
  }
  __syncthreads();
  const float* pl = xyz_large + ((size_t)b * NPT + n) * 3;
  float px = pl[0], py = pl[1], pz = pl[2];
  float d0 = 3.4e38f, d1 = 3.4e38f, d2 = 3.4e38f;
  int i0 = 0, i1 = 0, i2 = 0;
  for (int s = 0; s < SFEW; ++s) {
    float dx = px - fx[s], dy = py - fy[s], dz = pz - fz[s];
    float d = dx * dx + dy * dy + dz * dz;
    if (d < d2) {
      if (d < d1) {
        if (d < d0) { d2 = d1; i2 = i1; d1 = d0; i1 = i0; d0 = d; i0 = s; }
        else        { d2 = d1; i2 = i1; d1 = d;  i1 = s; }
      } else        { d2 = d;  i2 = s; }
    }
  }
  float r0 = 1.0f / (d0 + EPS_W), r1 = 1.0f / (d1 + EPS_W), r2 = 1.0f / (d2 + EPS_W);
  float rs = 1.0f / (r0 + r1 + r2);
  float w0 = r0 * rs, w1 = r1 * rs, w2 = r2 * rs;
  const float* F = feat_few + (size_t)b * SFEW * DFEW;
  const float* f0p = F + (size_t)i0 * DFEW;
  const float* f1p = F + (size_t)i1 * DFEW;
  const float* f2p = F + (size_t)i2 * DFEW;
  size_t col = (size_t)b * NPT + n;
  for (int c = 0; c < DFEW; ++c) {
    float v = w0 * f0p[c] + w1 * f1p[c] + w2 * f2p[c];
    X[(size_t)(DLRG + c) * NT + col] = v;
  }
}

// ---------------- transpose feat_large [col][c] -> X[c][col], rows 0..127 ----
__global__ __launch_bounds__(256) void k_transpose(
    const float* __restrict__ src, float* __restrict__ X) {
  __shared__ float tile[32][33];
  int c0 = blockIdx.x * 32;     // channel base (0..96)
  int col0 = blockIdx.y * 32;   // column base
  int tx = threadIdx.x & 31, ty = threadIdx.x >> 5;   // 32 x 8
  #pragma unroll
  for (int j = 0; j < 4; ++j) {
    int row = col0 + ty + 8 * j;
    tile[ty + 8 * j][tx] = src[(size_t)row * DLRG + c0 + tx];
  }
  __syncthreads();
  #pragma unroll
  for (int j = 0; j < 4; ++j) {
    int c = c0 + ty + 8 * j;
    X[(size_t)c * NT + col0 + tx] = tile[tx][ty + 8 * j];
  }
}

// ---------------- GEMM1: Y1 = W0*X + b0, accumulate BN stats --------------
// Block = 512 thr (16 waves, one 16-row M tile each); 2 N-sub-tiles per wave
// sharing the A fragment -> per K-step: 1 global b64 + 2 ds_load_2addr + 2 wmma.
__global__ __launch_bounds__(512) void k_gemm1(
    const float* __restrict__ W, const float* __restrict__ bias,
    const float* __restrict__ X, float* __restrict__ Y,
    float* __restrict__ sum, float* __restrict__ sumsq) {
  __shared__ float xt[2 * TS1];
  int col0 = blockIdx.x * 32;
  for (int i = threadIdx.x; i < CIN * 32; i += 512) {
    int k = i >> 5, c = i & 31;                     // 128B-coalesced read
    xt[(c >> 4) * TS1 + (k << 4) + (c & 15)] = X[(size_t)k * NT + col0 + c];
  }
  __syncthreads();
  int wave = threadIdx.x >> 5;
  int lane = threadIdx.x & 31;
  int mbase = wave * 16;                 // 16 waves cover 256 rows
  int mrow = mbase + (lane & 15);
  int koff = (lane >> 4) << 1;           // 0 or 2 (K pair per lane half)
  const float* wrow = W + (size_t)mrow * CIN;
  v8f acc0 = {}, acc1 = {};
  #pragma unroll 4
  for (int k = 0; k < CIN; k += 4) {
    v2f a, b0, b1;
    a.x = wrow[k + koff];
    a.y = wrow[k + koff + 1];
    b0.x = xt[(k + koff) * 16 + (lane & 15)];
    b0.y = xt[(k + koff + 1) * 16 + (lane & 15)];
    b1.x = xt[TS1 + (k + koff) * 16 + (lane & 15)];
    b1.y = xt[TS1 + (k + koff + 1) * 16 + (lane & 15)];
    acc0 = __builtin_amdgcn_wmma_f32_16x16x4_f32(false, a, false, b0, (short)0,
                                                 acc0, false, false);
    acc1 = __builtin_amdgcn_wmma_f32_16x16x4_f32(false, a, false, b1, (short)0,
                                                 acc1, false, false);
  }
  int half = (lane >> 4) << 3;           // lanes 16-31 hold rows +8
  #pragma unroll
  for (int v = 0; v < 8; ++v) {
    int row = mbase + half + v;
    float bv = bias[row];
    float y0 = acc0[v] + bv;
    float y1 = acc1[v] + bv;
    Y[(size_t)row * NT + col0 + (lane & 15)]      = y0;
    Y[(size_t)row * NT + col0 + 16 + (lane & 15)] = y1;
    float s = y0 + y1, s2 = y0 * y0 + y1 * y1;
    #pragma unroll
    for (int off = 1; off < 16; off <<= 1) {
      s  += __shfl_xor(s, off, 32);
      s2 += __shfl_xor(s2, off, 32);
    }
    if ((lane & 15) == 0) {
      atomicAdd(&sum[row], s);
      atomicAdd(&sumsq[row], s2);
    }
  }
}

// ---------------- BN parameter computation --------------------------------
__global__ void k_bnparams(const float* __restrict__ sum,
                           const float* __restrict__ sumsq,
                           const float* __restrict__ gamma,
                           const float* __restrict__ beta,
                           float* __restrict__ scale, float* __restrict__ shift,
                           int C, float inv_count) {
  int c = threadIdx.x;
  if (c < C) {
    float m = sum[c] * inv_count;
    float var = sumsq[c] * inv_count - m * m;
    float sc = gamma[c] * rsqrtf(var + EPS_BN);
    scale[c] = sc;
    shift[c] = beta[c] - m * sc;
  }
}

// ---------------- GEMM2: Y2 = W1*relu(bn(Y1)) + b1, accumulate stats ------
__global__ __launch_bounds__(256) void k_gemm2(
    const float* __restrict__ W, const float* __restrict__ bias,
    const float* __restrict__ Y1, const float* __restrict__ scale1,
    const float* __restrict__ shift1, float* __restrict__ Y2,
    float* __restrict__ sum, float* __restrict__ sumsq) {
  __shared__ float xt[2 * TS2];
  int col0 = blockIdx.x * 32;
  for (int i = threadIdx.x; i < C1 * 32; i += 256) {
    int k = i >> 5, c = i & 31;
    float v = fmaf(Y1[(size_t)k * NT + col0 + c], scale1[k], shift1[k]);
    xt[(c >> 4) * TS2 + (k << 4) + (c & 15)] = fmaxf(v, 0.0f);
  }
  __syncthreads();
  int wave = threadIdx.x >> 5;
  int lane = threadIdx.x & 31;
  int mbase = wave * 16;                 // 8 waves cover 128 rows
  int mrow = mbase + (lane & 15);
  int koff = (lane >> 4) << 1;
  const float* wrow = W + (size_t)mrow * C1;
  v8f acc0 = {}, acc1 = {};
  #pragma unroll 4
  for (int k = 0; k < C1; k += 4) {
    v2f a, b0, b1;
    a.x = wrow[k + koff];
    a.y = wrow[k + koff + 1];
    b0.x = xt[(k + koff) * 16 + (lane & 15)];
    b0.y = xt[(k + koff + 1) * 16 + (lane & 15)];
    b1.x = xt[TS2 + (k + koff) * 16 + (lane & 15)];
    b1.y = xt[TS2 + (k + koff + 1) * 16 + (lane & 15)];
    acc0 = __builtin_amdgcn_wmma_f32_16x16x4_f32(false, a, false, b0, (short)0,
                                                 acc0, false, false);
    acc1 = __builtin_amdgcn_wmma_f32_16x16x4_f32(false, a, false, b1, (short)0,
                                                 acc1, false, false);
  }
  int half = (lane >> 4) << 3;
  #pragma unroll
  for (int v = 0; v < 8; ++v) {
    int row = mbase + half + v;
    float bv = bias[row];
    float y0 = acc0[v] + bv;
    float y1 = acc1[v] + bv;
    Y2[(size_t)row * NT + col0 + (lane & 15)]      = y0;
    Y2[(size_t)row * NT + col0 + 16 + (lane & 15)] = y1;
    float s = y0 + y1, s2 = y0 * y0 + y1 * y1;
    #pragma unroll
    for (int off = 1; off < 16; off <<= 1) {
      s  += __shfl_xor(s, off, 32);
      s2 += __shfl_xor(s2, off, 32);
    }
    if ((lane & 15) == 0) {
      atomicAdd(&sum[row], s);
      atomicAdd(&sumsq[row], s2);
    }
  }
}

// ---------------- final BN2 + ReLU -> out[B][128][N] ----------------------
__global__ __launch_bounds__(256) void k_apply(
    const float* __restrict__ Y2, const float* __restrict__ scale,
    const float* __restrict__ shift, float* __restrict__ out) {
  size_t idx = (size_t)blockIdx.x * 256 + threadIdx.x;   // o*NT + col
  int o = (int)(idx >> 16);
  int col = (int)(idx & (NT - 1));
  int b = col >> 13;            // /8192
  int n = col & (NPT - 1);
  float v = Y2[idx] * scale[o] + shift[o];
  out[(((size_t)b * C2 + o) << 13) + n] = fmaxf(v, 0.0f);
}

extern "C" void kernel_launch(void* const* d_in, const int* in_sizes, int n_in,
                              void* d_out, int out_size, void* d_ws, size_t ws_size,
                              hipStream_t stream) {
  (void)in_sizes; (void)n_in; (void)out_size; (void)ws_size;
  const float* xyz_few   = (const float*)d_in[0];
  const float* feat_few  = (const float*)d_in[1];
  const float* xyz_large = (const float*)d_in[2];
  const float* feat_large= (const float*)d_in[3];
  const float* W0 = (const float*)d_in[4];
  const float* b0 = (const float*)d_in[5];
  const float* g0 = (const float*)d_in[6];
  const float* be0= (const float*)d_in[7];
  const float* W1 = (const float*)d_in[8];
  const float* b1 = (const float*)d_in[9];
  const float* g1 = (const float*)d_in[10];
  const float* be1= (const float*)d_in[11];

  float* ws = (float*)d_ws;
  float* X  = ws;                              // 384*65536
  float* Y1 = ws + (size_t)CIN * NT;           // 256*65536
  float* Y2 = Y1 + (size_t)C1 * NT;            // 128*65536
  float* st = Y2 + (size_t)C2 * NT;
  float* sum1 = st,        *sumsq1 = st + 256, *scale1 = st + 512,  *shift1 = st + 768;
  float* sum2 = st + 1024, *sumsq2 = st + 1152, *scale2 = st + 1280, *shift2 = st + 1408;

  k_zero<<<(1536 + 255) / 256, 256, 0, stream>>>(st, 1536);
  k_interp<<<dim3(NPT / 256, NB), 256, 0, stream>>>(xyz_few, feat_few, xyz_large, X);
  k_transpose<<<dim3(DLRG / 32, NT / 32), 256, 0, stream>>>(feat_large, X);
  k_gemm1<<<NT / 32, 512, 0, stream>>>(W0, b0, X, Y1, sum1, sumsq1);
  k_bnparams<<<1, 256, 0, stream>>>(sum1, sumsq1, g0, be0, scale1, shift1, C1, 1.0f / NT);
  k_gemm2<<<NT / 32, 256, 0, stream>>>(W1, b1, Y1, scale1, shift1, Y2, sum2, sumsq2);
  k_bnparams<<<1, 128, 0, stream>>>(sum2, sumsq2, g1, be1, scale2, shift2, C2, 1.0f / NT);
  k_apply<<<(size_t)C2 * NT / 256, 256, 0, stream>>>(Y2, scale2, shift2, (float*)d_out);
}